// CLSNet_18004502904989
// MI455X (gfx1250) — compile-verified
//
#include <hip/hip_runtime.h>
#include <math.h>

typedef float v2f __attribute__((ext_vector_type(2)));
typedef float v8f __attribute__((ext_vector_type(8)));

#define NC 6
#define NPROTO 48
#define CCH 128
#define HH 128
#define WW 128
#define NPIX 16384   // 128*128
#define BATCH 8

__device__ __forceinline__ float gelu_exact(float x) {
    return 0.5f * x * (1.0f + erff(x * 0.70710678118654752f));
}

// ---------------------------------------------------------------------------
// K1: adjusted label map: adjLab[b,h,w] = (gt[b,4h,4w]==6 ? 0 : gt)
// ---------------------------------------------------------------------------
__global__ __launch_bounds__(256) void k_adjlab(const long long* __restrict__ gt,
                                                int* __restrict__ adjLab) {
    int idx = blockIdx.x * 256 + threadIdx.x;            // 8*16384
    if (idx >= BATCH * NPIX) return;
    int b = idx >> 14;
    int r = idx & (NPIX - 1);
    int h = r >> 7, w = r & 127;
    long long v = gt[((size_t)b * 512 + 4 * h) * 512 + 4 * w];
    adjLab[idx] = (v == 6) ? 0 : (int)v;
}

// ---------------------------------------------------------------------------
// K2: per-patch context vectors -> layernorm(C) -> gelu -> centers (768x128)
// Replicates the reference's reshape-scrambled one_hot (pred0) exactly.
// One block per patch (128 patches), 128 threads = channels.
// ---------------------------------------------------------------------------
__global__ __launch_bounds__(128) void k_centers(const float* __restrict__ feats,
                                                 const int* __restrict__ adjLab,
                                                 const float* __restrict__ g1,
                                                 const float* __restrict__ b1,
                                                 float* __restrict__ centers) {
    const int bp = blockIdx.x;
    const int b = bp >> 4, pidx = bp & 15;
    const int pi = pidx >> 2, pj = pidx & 3;
    const int t = threadIdx.x;

    __shared__ unsigned char msk[1024];   // bit cls set -> pp[cls][p]==1
    __shared__ int cnt[NC];
    __shared__ float ctxs[NC][CCH];
    __shared__ float mu_s[NC], rs_s[NC];

    if (t < NC) cnt[t] = 0;
    __syncthreads();

    const int* alab = adjLab + b * NPIX;
    for (int p = t; p < 1024; p += 128) {
        int y = pi * 32 + (p >> 5);
        int x = pj * 32 + (p & 31);
        int base = y * 128 + x;
        unsigned m = 0;
#pragma unroll
        for (int cls = 0; cls < NC; ++cls) {
            int r  = cls * NPIX + base;          // flat index into pred0[b]
            int h  = r / 768;                    // Wh*NC = 768
            int rr = r - h * 768;
            int w  = rr / 6;
            int k  = rr - w * 6;
            if (alab[h * 128 + w] == k) {
                m |= 1u << cls;
                atomicAdd(&cnt[cls], 1);
            }
        }
        msk[p] = (unsigned char)m;
    }
    __syncthreads();

    // class-conditional feature sums for channel c = t
    const int c = t;
    const float* fp = feats + ((size_t)(b * CCH + c)) * NPIX + (pi * 32) * 128 + pj * 32;
    float s[NC] = {0.f, 0.f, 0.f, 0.f, 0.f, 0.f};
    for (int p = 0; p < 1024; ++p) {
        float f = fp[(p >> 5) * 128 + (p & 31)];
        unsigned m = msk[p];
#pragma unroll
        for (int cls = 0; cls < NC; ++cls)
            s[cls] += ((m >> cls) & 1u) ? f : 0.0f;
    }
    float sall = s[0] + s[1] + s[2] + s[3] + s[4] + s[5];

    const float E1 = 2.7182818284590452f;
#pragma unroll
    for (int cls = 0; cls < NC; ++cls) {
        float cf = (float)cnt[cls];
        float Z = cf * E1 + (1024.0f - cf);                 // softmax partition
        ctxs[cls][c] = (sall + (E1 - 1.0f) * s[cls]) / Z;   // ctx[k][c]
    }
    __syncthreads();

    if (t < NC) {
        float su = 0.f, sq = 0.f;
        for (int i = 0; i < CCH; ++i) {
            float v = ctxs[t][i];
            su += v; sq += v * v;
        }
        float mu = su * (1.0f / CCH);
        float var = sq * (1.0f / CCH) - mu * mu;
        mu_s[t] = mu;
        rs_s[t] = rsqrtf(var + 1e-6f);
    }
    __syncthreads();

#pragma unroll
    for (int cls = 0; cls < NC; ++cls) {
        float xv = (ctxs[cls][c] - mu_s[cls]) * rs_s[cls] * g1[c] + b1[c];
        centers[((size_t)(b * 16 + pidx) * NC + cls) * CCH + c] = gelu_exact(xv);
    }
}

// ---------------------------------------------------------------------------
// K3: nearest prototype per center (argmin of L2, first-min tiebreak)
// ---------------------------------------------------------------------------
__global__ __launch_bounds__(64) void k_nearest(const float* __restrict__ centers,
                                                const float* __restrict__ proto,
                                                int* __restrict__ nearest) {
    const int r = blockIdx.x;          // 768 centers
    const int t = threadIdx.x;
    __shared__ float crow[CCH];
    __shared__ float d2s[NPROTO];
    crow[t]      = centers[(size_t)r * CCH + t];
    crow[t + 64] = centers[(size_t)r * CCH + 64 + t];
    __syncthreads();
    if (t < NPROTO) {
        float acc = 0.f;
        const float* pr = proto + (size_t)t * CCH;
        for (int cc = 0; cc < CCH; ++cc) {
            float d = crow[cc] - pr[cc];
            acc = fmaf(d, d, acc);
        }
        d2s[t] = acc;
    }
    __syncthreads();
    if (t == 0) {
        float best = d2s[0]; int bi = 0;
        for (int k = 1; k < NPROTO; ++k)
            if (d2s[k] < best) { best = d2s[k]; bi = k; }
        nearest[r] = bi;
    }
}

// ---------------------------------------------------------------------------
// K4: sequential EMA scan over 768 (idx, center) pairs; emit proto_new,
// prototype_expand, and per-row squared norms p2.
// Single block, 128 threads (one per channel) -> each LDS cell owned by one
// thread, so the 768-step scan needs no barriers.
// ---------------------------------------------------------------------------
__global__ __launch_bounds__(128) void k_protoupdate(const float* __restrict__ proto_in,
                                                     const float* __restrict__ centers,
                                                     const int* __restrict__ nearest,
                                                     float* __restrict__ protoNew_ws,
                                                     float* __restrict__ p2_ws,
                                                     float* __restrict__ out_protoNew,
                                                     float* __restrict__ out_protoExpand) {
    __shared__ float P[NPROTO * CCH];
    __shared__ int idxs[768];
    const int t = threadIdx.x;
    for (int i = t; i < NPROTO * CCH; i += 128) P[i] = proto_in[i];
    for (int i = t; i < 768; i += 128) idxs[i] = nearest[i];
    __syncthreads();

    const float MOM = 0.999f;
    for (int i = 0; i < 768; ++i) {
        int idx = idxs[i];
        float v = centers[(size_t)i * CCH + t];
        float* pe = &P[idx * CCH + t];
        *pe = MOM * (*pe) + (1.0f - MOM) * v;
    }
    __syncthreads();

    for (int i = t; i < NPROTO * CCH; i += 128) {
        float v = P[i];
        protoNew_ws[i] = v;
        out_protoNew[i] = v;
#pragma unroll
        for (int b = 0; b < BATCH; ++b)
            out_protoExpand[(size_t)b * NPROTO * CCH + i] = v;
    }
    if (t < NPROTO) {
        float a = 0.f;
        for (int cc = 0; cc < CCH; ++cc) a = fmaf(P[t * CCH + cc], P[t * CCH + cc], a);
        p2_ws[t] = a;
    }
}

// ---------------------------------------------------------------------------
// K5: main fused kernel. Per wave: one 16-pixel tile x all 48 prototypes via
// 3 accumulators of V_WMMA_F32_16X16X4_F32 over K=128. Fused epilogue:
// distance_l2, 1/(1+2d), layernorm over 48 (cross half-wave shfl), gelu,
// p2c_sim_map, and max-over-KP pred map.
// PPITCH=132 (== 4 mod 64): the 32 lanes' ds_load_b64 A-fragment reads tile
// all 64 LDS banks exactly once -> conflict-free.
// ---------------------------------------------------------------------------
#define PPITCH 132

__global__ __launch_bounds__(256) void k_distance(const float* __restrict__ feats,
                                                  const float* __restrict__ protoNew,
                                                  const float* __restrict__ p2w,
                                                  const float* __restrict__ g2,
                                                  const float* __restrict__ b2,
                                                  float* __restrict__ outD,
                                                  float* __restrict__ outP,
                                                  float* __restrict__ predSmall) {
    __shared__ float Pl[NPROTO * PPITCH];
    __shared__ float p2l[NPROTO], g2l[NPROTO], b2l[NPROTO];

    const int tid = threadIdx.x;
    for (int i = tid; i < NPROTO * CCH; i += 256)
        Pl[(i >> 7) * PPITCH + (i & 127)] = protoNew[i];
    if (tid < NPROTO) { p2l[tid] = p2w[tid]; g2l[tid] = g2[tid]; b2l[tid] = b2[tid]; }
    __syncthreads();

    const int wid  = tid >> 5;
    const int lane = tid & 31;
    const int hi   = lane >> 4;       // half-wave: K rows 2hi,2hi+1 / M rows +8
    const int nl   = lane & 15;
    const int b    = blockIdx.y;
    const int n0   = (blockIdx.x * 8 + wid) * 16;
    const int n    = n0 + nl;

    const float* fb = feats + (size_t)b * CCH * NPIX;

    v8f a0 = {0.f,0.f,0.f,0.f,0.f,0.f,0.f,0.f};
    v8f a1 = a0, a2 = a0;
    float f2p = 0.f;

#pragma unroll 8
    for (int c0 = 0; c0 < CCH; c0 += 4) {
        int cch = c0 + (hi << 1);
        size_t o = (size_t)cch * NPIX + n;
        float bx = fb[o];
        float by = fb[o + NPIX];
        if (c0 + 8 < CCH)
            __builtin_prefetch(&fb[o + (size_t)8 * NPIX], 0, 3);  // WGP-scope
        f2p = fmaf(bx, bx, fmaf(by, by, f2p));
        v2f bf; bf.x = bx; bf.y = by;
        v2f af0 = *(const v2f*)&Pl[(0  + nl) * PPITCH + cch];
        v2f af1 = *(const v2f*)&Pl[(16 + nl) * PPITCH + cch];
        v2f af2 = *(const v2f*)&Pl[(32 + nl) * PPITCH + cch];
        a0 = __builtin_amdgcn_wmma_f32_16x16x4_f32(false, af0, false, bf, (short)0, a0, false, false);
        a1 = __builtin_amdgcn_wmma_f32_16x16x4_f32(false, af1, false, bf, (short)0, a1, false, false);
        a2 = __builtin_amdgcn_wmma_f32_16x16x4_f32(false, af2, false, bf, (short)0, a2, false, false);
    }

    float f2 = f2p + __shfl_xor(f2p, 16, 32);   // full sum of squares for pixel n

    v8f accs[3] = {a0, a1, a2};
    float vals[24];
    float sum = 0.f, sq = 0.f;
#pragma unroll
    for (int t3 = 0; t3 < 3; ++t3) {
#pragma unroll
        for (int j = 0; j < 8; ++j) {
            int row = t3 * 16 + j + (hi << 3);
            float dot = accs[t3][j];
            float d2v = p2l[row] + f2 - 2.0f * dot;
            float dv = sqrtf(fmaxf(d2v, 0.0f));
            outD[((size_t)(b * NPROTO + row)) * NPIX + n] = dv;
            float vv = 1.0f / (1.0f + 2.0f * dv);
            vals[t3 * 8 + j] = vv;
            sum += vv;
            sq = fmaf(vv, vv, sq);
        }
    }
    sum += __shfl_xor(sum, 16, 32);
    sq  += __shfl_xor(sq, 16, 32);
    float mu  = sum * (1.0f / NPROTO);
    float var = sq * (1.0f / NPROTO) - mu * mu;
    float rstd = rsqrtf(var + 1e-6f);

    float pm[NC] = {-1e30f, -1e30f, -1e30f, -1e30f, -1e30f, -1e30f};
#pragma unroll
    for (int t3 = 0; t3 < 3; ++t3) {
#pragma unroll
        for (int j = 0; j < 8; ++j) {
            int row = t3 * 16 + j + (hi << 3);
            float xv = (vals[t3 * 8 + j] - mu) * rstd * g2l[row] + b2l[row];
            float yv = gelu_exact(xv);
            outP[((size_t)(b * NPROTO + row)) * NPIX + n] = yv;
            int cls = row % NC;
#pragma unroll
            for (int cl = 0; cl < NC; ++cl)
                pm[cl] = (cls == cl) ? fmaxf(pm[cl], yv) : pm[cl];
        }
    }
#pragma unroll
    for (int cl = 0; cl < NC; ++cl)
        pm[cl] = fmaxf(pm[cl], __shfl_xor(pm[cl], 16, 32));
    if (hi == 0) {
#pragma unroll
        for (int cl = 0; cl < NC; ++cl)
            predSmall[((size_t)(b * NC + cl)) * NPIX + n] = pm[cl];
    }
}

// ---------------------------------------------------------------------------
// K6: jax-style half-pixel bilinear resize (B,6,128,128) -> (B,6,512,512)
// ---------------------------------------------------------------------------
__global__ __launch_bounds__(256) void k_resize(const float* __restrict__ ps,
                                                float* __restrict__ pred) {
    int idx = blockIdx.x * 256 + threadIdx.x;     // 8*6*512*512
    if (idx >= BATCH * NC * 512 * 512) return;
    int x  = idx & 511;
    int y  = (idx >> 9) & 511;
    int bc = idx >> 18;

    float sx = (x + 0.5f) * 0.25f - 0.5f;
    float sy = (y + 0.5f) * 0.25f - 0.5f;
    int x0 = (int)floorf(sx), y0 = (int)floorf(sy);
    float wx = sx - (float)x0, wy = sy - (float)y0;
    int x0c = min(max(x0, 0), 127), x1c = min(max(x0 + 1, 0), 127);
    int y0c = min(max(y0, 0), 127), y1c = min(max(y0 + 1, 0), 127);

    const float* src = ps + (size_t)bc * NPIX;
    float v00 = src[y0c * 128 + x0c], v01 = src[y0c * 128 + x1c];
    float v10 = src[y1c * 128 + x0c], v11 = src[y1c * 128 + x1c];
    float v = (1.f - wy) * ((1.f - wx) * v00 + wx * v01)
            +        wy  * ((1.f - wx) * v10 + wx * v11);
    pred[idx] = v;
}

// ---------------------------------------------------------------------------
extern "C" void kernel_launch(void* const* d_in, const int* in_sizes, int n_in,
                              void* d_out, int out_size, void* d_ws, size_t ws_size,
                              hipStream_t stream) {
    (void)in_sizes; (void)n_in; (void)out_size; (void)ws_size;
    const float*     feats = (const float*)d_in[0];
    const long long* gt    = (const long long*)d_in[1];
    const float*     proto = (const float*)d_in[2];
    const float*     g1    = (const float*)d_in[3];
    const float*     b1    = (const float*)d_in[4];
    const float*     g2    = (const float*)d_in[5];
    const float*     b2    = (const float*)d_in[6];

    // d_out regions (tuple concat order)
    float* out = (float*)d_out;
    float* outPred      = out;                         // 8*6*512*512 = 12582912
    float* outProtoExp  = out + 12582912;              // 8*48*128    = 49152
    float* outP2C       = out + 12632064;              // 8*48*16384  = 6291456
    float* outDist      = out + 18923520;              // 8*48*16384  = 6291456
    float* outProtoNew  = out + 25214976;              // 48*128      = 6144

    // workspace carve
    float* w = (float*)d_ws;
    float* predSmall = w;                 // 786432
    float* centers   = w + 786432;        // 98304
    float* protoNew  = w + 884736;        // 6144
    float* p2w       = w + 890880;        // 48
    int*   nearest   = (int*)(w + 890928);   // 768
    int*   adjLab    = (int*)(w + 891696);   // 131072

    k_adjlab<<<(BATCH * NPIX + 255) / 256, 256, 0, stream>>>(gt, adjLab);
    k_centers<<<BATCH * 16, 128, 0, stream>>>(feats, adjLab, g1, b1, centers);
    k_nearest<<<768, 64, 0, stream>>>(centers, proto, nearest);
    k_protoupdate<<<1, 128, 0, stream>>>(proto, centers, nearest,
                                         protoNew, p2w, outProtoNew, outProtoExp);
    dim3 gdist(NPIX / (8 * 16), BATCH);   // (128, 8)
    k_distance<<<gdist, 256, 0, stream>>>(feats, protoNew, p2w, g2, b2,
                                          outDist, outP2C, predSmall);
    k_resize<<<(BATCH * NC * 512 * 512 + 255) / 256, 256, 0, stream>>>(predSmall, outPred);
}